// Set2Set_9277129360143
// MI455X (gfx1250) — compile-verified
//
#include <hip/hip_runtime.h>
#include <math.h>

#define NN 400000
#define CC 128
#define BB 1024
#define STEPS 3

typedef float v2f __attribute__((ext_vector_type(2)));
typedef float v8f __attribute__((ext_vector_type(8)));

// ---------------------------------------------------------------------------
// gates = q_star @ w_ih^T + h @ w_hh^T + (b_ih + b_hh)      [B, 4C]
// One wave per 16x16 tile of the [1024, 512] output, fp32 WMMA, K step = 4.
// A 16x4 layout: M = lane%16; lanes 0-15 -> K={k,k+1}, lanes 16-31 -> K={k+2,k+3}
// B 4x16 layout: N = lane%16; same K split (B row = W[n][k] since gates = q@W^T)
// D layout: VGPR i -> row m0 + i + 8*(lane>>4), col n0 + lane%16
// ---------------------------------------------------------------------------
__global__ __launch_bounds__(256) void gates_gemm(
    const float* __restrict__ qstar, const float* __restrict__ hmat,
    const float* __restrict__ w_ih, const float* __restrict__ w_hh,
    const float* __restrict__ b_ih, const float* __restrict__ b_hh,
    float* __restrict__ gates)
{
    const int tid   = threadIdx.x;
    const int wave  = tid >> 5;
    const int lane  = tid & 31;
    const int tile  = blockIdx.x * 8 + wave;      // 2048 tiles total
    const int m0    = (tile >> 5) << 4;           // 64 row tiles
    const int n0    = (tile & 31) << 4;           // 32 col tiles
    const int lrow  = lane & 15;
    const int khalf = lane >> 4;

    v8f acc = {0.f, 0.f, 0.f, 0.f, 0.f, 0.f, 0.f, 0.f};

    // ---- K = 0..255 over q_star / w_ih ----
    const float* arow = qstar + (size_t)(m0 + lrow) * (2 * CC);
    const float* brow = w_ih  + (size_t)(n0 + lrow) * (2 * CC);
    for (int k = 0; k < 2 * CC; k += 4) {
        v2f a = *(const v2f*)(arow + k + 2 * khalf);
        v2f b = *(const v2f*)(brow + k + 2 * khalf);
        acc = __builtin_amdgcn_wmma_f32_16x16x4_f32(false, a, false, b,
                                                    (short)0, acc, false, false);
    }
    // ---- K = 0..127 over h / w_hh ----
    const float* arow2 = hmat + (size_t)(m0 + lrow) * CC;
    const float* brow2 = w_hh + (size_t)(n0 + lrow) * CC;
    for (int k = 0; k < CC; k += 4) {
        v2f a = *(const v2f*)(arow2 + k + 2 * khalf);
        v2f b = *(const v2f*)(brow2 + k + 2 * khalf);
        acc = __builtin_amdgcn_wmma_f32_16x16x4_f32(false, a, false, b,
                                                    (short)0, acc, false, false);
    }

    const float bias = b_ih[n0 + lrow] + b_hh[n0 + lrow];  // same col for all 8 VGPRs
#pragma unroll
    for (int i = 0; i < 8; ++i) {
        const int row = m0 + i + 8 * khalf;
        gates[(size_t)row * (4 * CC) + n0 + lrow] = acc[i] + bias;
    }
}

// ---------------------------------------------------------------------------
// c = sigmoid(f)*c + sigmoid(i)*tanh(g);  h = sigmoid(o)*tanh(c)
// ---------------------------------------------------------------------------
__global__ __launch_bounds__(256) void lstm_pointwise(
    const float* __restrict__ gates, float* __restrict__ c, float* __restrict__ h)
{
    const int idx = blockIdx.x * 256 + threadIdx.x;   // < B*C
    const int b   = idx >> 7;
    const int ch  = idx & (CC - 1);
    const float* g = gates + (size_t)b * 4 * CC;
    const float gi = g[ch];
    const float gf = g[CC + ch];
    const float gg = g[2 * CC + ch];
    const float go = g[3 * CC + ch];
    const float si = 1.f / (1.f + expf(-gi));
    const float sf = 1.f / (1.f + expf(-gf));
    const float so = 1.f / (1.f + expf(-go));
    const float cn = sf * c[idx] + si * tanhf(gg);
    c[idx] = cn;
    h[idx] = so * tanhf(cn);
}

// ---------------------------------------------------------------------------
// Single-pass segment softmax attention. One block (8 waves) per graph.
// Lane l owns channels 4l..4l+3; each wave keeps 4 independent online
// (m, s, racc) states for ILP; 32 states merged via LDS with exact-max rescale.
// Writes q_star[b] = [ h[b], r[b] ].
// ---------------------------------------------------------------------------
__global__ __launch_bounds__(256) void set2set_attn(
    const float* __restrict__ x, const int* __restrict__ bidx,
    const float* __restrict__ h, float* __restrict__ qdst)
{
    __shared__ float sm_m[32];
    __shared__ float sm_s[32];
    __shared__ float sm_r[32 * CC];
    __shared__ int   sm_se[2];

    const int b   = blockIdx.x;
    const int tid = threadIdx.x;

    if (tid == 0) {
        int lo = 0, hi = NN;
        while (lo < hi) { int mid = (lo + hi) >> 1; if (bidx[mid] <  b) lo = mid + 1; else hi = mid; }
        sm_se[0] = lo;
        int lo2 = lo; hi = NN;
        while (lo2 < hi) { int mid = (lo2 + hi) >> 1; if (bidx[mid] <= b) lo2 = mid + 1; else hi = mid; }
        sm_se[1] = lo2;
    }
    __syncthreads();
    const int start = sm_se[0];
    const int end   = sm_se[1];

    const int wave = tid >> 5;
    const int lane = tid & 31;
    const float4 qv = *(const float4*)(h + (size_t)b * CC + 4 * lane);

    float  m[4], s[4];
    float4 r[4];
#pragma unroll
    for (int j = 0; j < 4; ++j) { m[j] = -INFINITY; s[j] = 0.f; r[j] = make_float4(0.f, 0.f, 0.f, 0.f); }

    for (int n0 = start + wave * 4; n0 < end; n0 += 32) {
        if (n0 + 32 < end)  // stream-ahead prefetch (global_prefetch_b8)
            __builtin_prefetch(x + (size_t)(n0 + 32) * CC + 4 * lane, 0, 0);

        float4 xv[4];
#pragma unroll
        for (int j = 0; j < 4; ++j)
            if (n0 + j < end)
                xv[j] = *(const float4*)(x + (size_t)(n0 + j) * CC + 4 * lane);

#pragma unroll
        for (int j = 0; j < 4; ++j) {
            if (n0 + j < end) {   // uniform across the wave
                float e = fmaf(xv[j].x, qv.x,
                          fmaf(xv[j].y, qv.y,
                          fmaf(xv[j].z, qv.z, xv[j].w * qv.w)));
                e += __shfl_xor(e, 16);
                e += __shfl_xor(e, 8);
                e += __shfl_xor(e, 4);
                e += __shfl_xor(e, 2);
                e += __shfl_xor(e, 1);
                const float nm   = fmaxf(m[j], e);
                const float corr = expf(m[j] - nm);   // exp(-inf)=0 on first node
                const float p    = expf(e - nm);
                s[j]   = s[j] * corr + p;
                r[j].x = r[j].x * corr + p * xv[j].x;
                r[j].y = r[j].y * corr + p * xv[j].y;
                r[j].z = r[j].z * corr + p * xv[j].z;
                r[j].w = r[j].w * corr + p * xv[j].w;
                m[j] = nm;
            }
        }
    }

#pragma unroll
    for (int j = 0; j < 4; ++j) {
        const int sid = wave * 4 + j;
        if (lane == 0) { sm_m[sid] = m[j]; sm_s[sid] = s[j]; }
        *(float4*)(sm_r + sid * CC + 4 * lane) = r[j];
    }
    __syncthreads();

    if (tid < CC) {
        float M = -INFINITY;
#pragma unroll
        for (int i = 0; i < 32; ++i) M = fmaxf(M, sm_m[i]);
        float S = 0.f, rs = 0.f;
        if (M > -INFINITY) {
            for (int i = 0; i < 32; ++i) {
                const float wgt = expf(sm_m[i] - M);   // 0 for empty states
                S  += wgt * sm_s[i];
                rs += wgt * sm_r[i * CC + tid];
            }
        }
        qdst[(size_t)b * 2 * CC + tid]      = h[(size_t)b * CC + tid];
        qdst[(size_t)b * 2 * CC + CC + tid] = rs / (S + 1e-16f);
    }
}

// ---------------------------------------------------------------------------
extern "C" void kernel_launch(void* const* d_in, const int* in_sizes, int n_in,
                              void* d_out, int out_size, void* d_ws, size_t ws_size,
                              hipStream_t stream)
{
    const float* x    = (const float*)d_in[0];
    const float* w_ih = (const float*)d_in[1];
    const float* w_hh = (const float*)d_in[2];
    const float* b_ih = (const float*)d_in[3];
    const float* b_hh = (const float*)d_in[4];
    const int*   bidx = (const int*)d_in[5];

    float* ws    = (float*)d_ws;
    float* qstar = ws;                       // B * 2C
    float* hbuf  = qstar + (size_t)BB * 2 * CC;  // B * C
    float* cbuf  = hbuf  + (size_t)BB * CC;      // B * C
    float* gates = cbuf  + (size_t)BB * CC;      // B * 4C

    // q_star, h, c start at zero every call (deterministic under graph replay)
    hipMemsetAsync(ws, 0, (size_t)(BB * 2 * CC + 2 * BB * CC) * sizeof(float), stream);

    for (int step = 0; step < STEPS; ++step) {
        gates_gemm<<<256, 256, 0, stream>>>(qstar, hbuf, w_ih, w_hh, b_ih, b_hh, gates);
        lstm_pointwise<<<(BB * CC) / 256, 256, 0, stream>>>(gates, cbuf, hbuf);
        float* qdst = (step == STEPS - 1) ? (float*)d_out : qstar;
        set2set_attn<<<BB, 256, 0, stream>>>(x, bidx, hbuf, qdst);
    }
}